// PropLoss_82978768159398
// MI455X (gfx1250) — compile-verified
//
#include <hip/hip_runtime.h>

typedef __attribute__((ext_vector_type(2))) float v2f;
typedef __attribute__((ext_vector_type(8))) float v8f;

#define F_DIM 16

// ---------------------------------------------------------------------------
// Kernel 1: zero the Ad accumulator (in d_ws) and the scalar output.
// float4 stores -> global_store_b128, 4x fewer vmem ops for the 6.4 MB clear.
// nfeat = N*16 is always divisible by 4.
// ---------------------------------------------------------------------------
__global__ void prop_zero_kernel(float4* __restrict__ ad4, long long n4,
                                 float* __restrict__ out) {
    long long i = (long long)blockIdx.x * blockDim.x + threadIdx.x;
    long long stride = (long long)gridDim.x * blockDim.x;
    float4 z; z.x = 0.0f; z.y = 0.0f; z.z = 0.0f; z.w = 0.0f;
    for (; i < n4; i += stride) ad4[i] = z;
    if (blockIdx.x == 0 && threadIdx.x == 0) out[0] = 0.0f;
}

// ---------------------------------------------------------------------------
// Kernel 2: edge scatter. 16 lanes cooperate on one edge -> coalesced 64B
// gather of d[src] and coalesced 64B burst of f32 atomics to Ad[dst].
// Scalar edge loads (src/dst/val/mask) are same-address within each 16-lane
// group -> a single coalesced request per wave per stream.
// d (6.4 MB) and Ad (6.4 MB) are L2-resident (192 MB L2), so the 51.2M
// no-return global_atomic_add_f32 resolve at L2 — the true bottleneck.
// ---------------------------------------------------------------------------
__global__ void prop_scatter_kernel(const float* __restrict__ d,
                                    const long long* __restrict__ ei,
                                    const float* __restrict__ vals,
                                    const unsigned char* __restrict__ mask,
                                    float* __restrict__ ad,
                                    long long E) {
    const int f = threadIdx.x & (F_DIM - 1);
    long long gid = (long long)blockIdx.x * blockDim.x + threadIdx.x;
    long long group = gid >> 4;                                  // 16 lanes / edge
    long long ngroups = ((long long)gridDim.x * blockDim.x) >> 4;
    for (long long e = group; e < E; e += ngroups) {
        long long src = ei[e];
        long long dst = ei[E + e];
        float v = mask[e] ? vals[e] : 0.0f;
        float contrib = v * d[src * F_DIM + f];
        unsafeAtomicAdd(&ad[dst * F_DIM + f], contrib);          // global_atomic_add_f32 (no-return)
    }
}

// ---------------------------------------------------------------------------
// Kernel 3: MSE reduction. float4 streaming loads (global_load_b128) of Ad
// and residual, then an exact f32 wave32 reduction via V_WMMA_F32_16X16X4_F32:
//
//   A (16x4 f32, documented layout): lane l<16 holds A[l][0..1], lane l>=16
//   holds A[l-16][2..3]. Both A components = lane partial p gives row m =
//   {p_m, p_m, p_{m+16}, p_{m+16}}. B = all-ones, so
//   D[m][n] = sum_k A[m][k] = 2*(p_m + p_{m+16}) for every n.
//   Lane-local sum of the 8 D VGPRs covers rows {0..7} (lanes 0-15) or
//   {8..15} (lanes 16-31); one shfl_xor(16) + *0.5 gives the exact 32-lane
//   sum. EXEC is all-ones at the WMMA (full reconvergence, no early return).
// ---------------------------------------------------------------------------
__global__ void prop_mse_kernel(const float4* __restrict__ ad4,
                                const float4* __restrict__ res4,
                                float* __restrict__ out,
                                long long n4, float inv_total) {
    float p = 0.0f;
    long long i = (long long)blockIdx.x * blockDim.x + threadIdx.x;
    long long stride = (long long)gridDim.x * blockDim.x;
    for (; i < n4; i += stride) {
        float4 a = ad4[i];
        float4 r = res4[i];
        float e0 = a.x - r.x;
        float e1 = a.y - r.y;
        float e2 = a.z - r.z;
        float e3 = a.w - r.w;
        p = fmaf(e0, e0, p);
        p = fmaf(e1, e1, p);
        p = fmaf(e2, e2, p);
        p = fmaf(e3, e3, p);
    }

    v2f a2; a2.x = p;    a2.y = p;
    v2f b2; b2.x = 1.0f; b2.y = 1.0f;
    v8f cacc = {};
    v8f dmat = __builtin_amdgcn_wmma_f32_16x16x4_f32(
        /*neg_a=*/false, a2, /*neg_b=*/false, b2,
        /*c_mod=*/(short)0, cacc, /*reuse_a=*/false, /*reuse_b=*/false);

    float s = dmat[0] + dmat[1] + dmat[2] + dmat[3] +
              dmat[4] + dmat[5] + dmat[6] + dmat[7];
    s += __shfl_xor(s, 16, 32);
    float wave_sum = 0.5f * s;

    if ((threadIdx.x & 31) == 0)
        unsafeAtomicAdd(out, wave_sum * inv_total);
}

// ---------------------------------------------------------------------------
// Launch. Inputs (setup_inputs order):
//   d_in[0] = d              [N*16]  f32
//   d_in[1] = edge_index     [2*E]   i64
//   d_in[2] = matrix_values  [E]     f32
//   d_in[3] = mask           [E]     bool(u8)
//   d_in[4] = residual       [N*16]  f32
// d_out: 1 float. d_ws: Ad accumulator (N*16 f32 = 6.4 MB).
// ---------------------------------------------------------------------------
extern "C" void kernel_launch(void* const* d_in, const int* in_sizes, int n_in,
                              void* d_out, int out_size, void* d_ws, size_t ws_size,
                              hipStream_t stream) {
    const float* d             = (const float*)d_in[0];
    const long long* ei        = (const long long*)d_in[1];
    const float* vals          = (const float*)d_in[2];
    const unsigned char* mask  = (const unsigned char*)d_in[3];
    const float4* res4         = (const float4*)d_in[4];
    float* out = (float*)d_out;
    float* ad  = (float*)d_ws;

    long long nfeat = (long long)in_sizes[0];   // N * 16
    long long n4    = nfeat >> 2;               // float4 elements
    long long E     = (long long)in_sizes[1] / 2;

    prop_zero_kernel<<<1024, 256, 0, stream>>>((float4*)ad, n4, out);
    prop_scatter_kernel<<<8192, 256, 0, stream>>>(d, ei, vals, mask, ad, E);
    prop_mse_kernel<<<1024, 256, 0, stream>>>((const float4*)ad, res4, out, n4,
                                              1.0f / (float)nfeat);
}